// OrthogonalIntervention_55774445306383
// MI455X (gfx1250) — compile-verified
//
#include <hip/hip_runtime.h>
#include <cmath>

// OrthogonalIntervention for MI455X (gfx1250, wave32).
// Memory-bound op (~512 MB HBM traffic -> ~22 us at 23.3 TB/s):
//   K1: fused copy out=h + per-token ||h||^2 (single read+write of h)
//   K2: per-batch top-k (k = max(1, floor(n_vis*0.01)) <= 40) selection
//   K3: rank-4 steering of only the selected tokens via v_wmma_f32_16x16x4_f32

typedef __attribute__((ext_vector_type(2))) float v2f;
typedef __attribute__((ext_vector_type(8))) float v8f;

#define B_ 4
#define S_ 4096
#define D_ 4096
#define MAXK 64
#define TILES 4 // MAXK/16

typedef unsigned char MaskT; // numpy bool -> 1 byte

// ---------------------------------------------------------------------------
// Kernel 1: fused streaming copy out = h and per-token squared L2 norm.
// One wave per token, float4 traffic, wave32 shuffle reduction.
// ---------------------------------------------------------------------------
__global__ __launch_bounds__(256) void oi_norm_copy(const float* __restrict__ h,
                                                    float* __restrict__ out,
                                                    float* __restrict__ norms2)
{
    const int wave = threadIdx.x >> 5;
    const int lane = threadIdx.x & 31;
    const size_t tkn = (size_t)blockIdx.x * 8 + wave; // 8 waves per block
    const float4* __restrict__ src = (const float4*)(h + tkn * (size_t)D_);
    float4* __restrict__ dst = (float4*)(out + tkn * (size_t)D_);
    float s = 0.0f;
#pragma unroll 4
    for (int it = 0; it < D_ / 4 / 32; ++it) {
        float4 v = src[it * 32 + lane];
        dst[it * 32 + lane] = v;
        s += v.x * v.x + v.y * v.y + v.z * v.z + v.w * v.w;
    }
#pragma unroll
    for (int off = 16; off; off >>= 1) s += __shfl_xor(s, off, 32);
    if (lane == 0) norms2[tkn] = s;
}

// ---------------------------------------------------------------------------
// Kernel 2: per-batch top-k selection by norm^2 among visual tokens.
// Tie-break = smaller index first (matches reference's stable argsort).
// ---------------------------------------------------------------------------
__global__ __launch_bounds__(256) void oi_select(const float* __restrict__ norms2,
                                                 const MaskT* __restrict__ vis,
                                                 int* __restrict__ steer_idx,
                                                 int* __restrict__ steer_cnt)
{
    __shared__ int s_int[256];
    __shared__ unsigned long long s_key[256];
    __shared__ unsigned char sel[S_];
    const int b = blockIdx.x;
    const int tid = threadIdx.x;

    for (int i = tid; i < S_; i += 256) sel[i] = 0;

    int c = 0;
    for (int i = tid; i < S_; i += 256) c += vis[b * S_ + i] ? 1 : 0;
    s_int[tid] = c;
    __syncthreads();
    for (int off = 128; off; off >>= 1) {
        if (tid < off) s_int[tid] += s_int[tid + off];
        __syncthreads();
    }
    const int n_vis = s_int[0];

    int k = 0;
    if (n_vis > 0) {
        k = (int)floorf((float)n_vis * 0.01f);
        if (k < 1) k = 1;
        if (k > MAXK) k = MAXK;
    }

    for (int i = 0; i < k; ++i) {
        // key = (norm2 bits desc, index asc); norm2 >= 0 so IEEE bit pattern
        // orders identically to the value.
        unsigned long long best = 0ull;
        for (int j = tid; j < S_; j += 256) {
            if (vis[b * S_ + j] && !sel[j]) {
                unsigned int fb = __float_as_uint(norms2[b * S_ + j]);
                unsigned long long key =
                    ((unsigned long long)fb << 32) |
                    (unsigned long long)(0xFFFFFFFFu - (unsigned)j);
                if (key > best) best = key;
            }
        }
        s_key[tid] = best;
        __syncthreads();
        for (int off = 128; off; off >>= 1) {
            if (tid < off) {
                if (s_key[tid + off] > s_key[tid]) s_key[tid] = s_key[tid + off];
            }
            __syncthreads();
        }
        if (tid == 0) {
            int idx = (int)(0xFFFFFFFFu - (unsigned)(s_key[0] & 0xFFFFFFFFull));
            sel[idx] = 1;
            steer_idx[b * MAXK + i] = idx;
        }
        __syncthreads();
    }
    if (tid == 0) steer_cnt[b] = k;
}

// ---------------------------------------------------------------------------
// Kernel 3: WMMA steering of selected tokens. One wave per 16-token tile.
//
// Phase 1: acc[16x16] += A(16x4 h-slice) x B(4x16) over K, B columns
//   0..3 = R rows, 4..7 = W_w rows. Columns 8..15 and padded A rows carry
//   garbage by design: WMMA row/column independence means they never
//   contaminate the columns/rows we consume, so ALL loads are unconditional
//   (no EXEC toggling in the hot loop; WMMA sees EXEC = all-1s throughout).
// Phase 2: per 16-col chunk of D: D = delta_low(16x4) x R-chunk(4x16) + C,
//   C preloaded from h -> D is directly the output rows. Only the final
//   stores are masked (padded rows alias token 0 and must not write).
//
// f32 WMMA operand layouts (ISA 7.12.2):
//   A 16x4: lane = M%16, lane-half selects K pair.
//   B 4x16: lane = N%16, lane-half selects K pair.
//   C/D 16x16: VGPR j, lane L -> (M = j + 8*(L>=16), N = L%16).
// ---------------------------------------------------------------------------
__global__ __launch_bounds__(32) void oi_steer_wmma(const float* __restrict__ h,
                                                    const float* __restrict__ Rm,
                                                    const float* __restrict__ Ww,
                                                    const float* __restrict__ Wb,
                                                    const int* __restrict__ steer_idx,
                                                    const int* __restrict__ steer_cnt,
                                                    float* __restrict__ out)
{
    const int bb = blockIdx.x >> 2;
    const int tile = blockIdx.x & 3;
    int cnt = steer_cnt[bb] - tile * 16;
    if (cnt <= 0) return;
    if (cnt > 16) cnt = 16;

    __shared__ int tok[16];
    __shared__ float rw[16][8]; // [row][0..3]=Rh, [4..7]=Wh
    __shared__ float dl[16][4]; // delta_low

    const int lane = threadIdx.x;
    if (lane < 16) tok[lane] = (lane < cnt) ? steer_idx[bb * MAXK + tile * 16 + lane] : 0;
    __syncthreads();

    const int m = lane & 15;  // M (phase-1 A) / N (B, C, D)
    const int kh = lane >> 4; // lane half -> K pair
    const float* __restrict__ hb = h + (size_t)bb * S_ * D_;
    const float* __restrict__ arow = hb + (size_t)tok[m] * D_;
    // B source row for column n=m: R rows for n<4, W rows for 4<=n<8.
    // Columns 8..15 read R row 0 -- results in those columns are ignored.
    const float* __restrict__ bsrc =
        (m < 4) ? (Rm + (size_t)m * D_)
                : ((m < 8) ? (Ww + (size_t)(m - 4) * D_) : Rm);

    v8f acc = {0.f, 0.f, 0.f, 0.f, 0.f, 0.f, 0.f, 0.f};
#pragma unroll 8
    for (int kb = 0; kb < D_ / 4; ++kb) {
        const int d0 = kb * 4 + kh * 2; // even float index -> 8B aligned
        v2f A  = *(const v2f*)(arow + d0);
        v2f Bv = *(const v2f*)(bsrc + d0);
        acc = __builtin_amdgcn_wmma_f32_16x16x4_f32(false, A, false, Bv,
                                                    (short)0, acc, false, false);
    }

    // Spill the 16x8 useful part of acc (N = m < 8) to LDS.
#pragma unroll
    for (int j = 0; j < 8; ++j) {
        const int mm = j + kh * 8;
        if (m < 8) rw[mm][m] = acc[j];
    }
    __syncthreads();

    // delta_low = Wh + b - Rh; zero for padded rows so they contribute 0.
    for (int q = lane; q < 64; q += 32) {
        const int mm = q >> 2, jj = q & 3;
        dl[mm][jj] = (mm < cnt) ? (rw[mm][4 + jj] + Wb[jj] - rw[mm][jj]) : 0.0f;
    }
    __syncthreads();

    v2f A2;
    A2.x = dl[m][kh * 2];
    A2.y = dl[m][kh * 2 + 1];
    const size_t outb = (size_t)bb * S_ * D_;

#pragma unroll 2
    for (int cblk = 0; cblk < D_ / 16; ++cblk) {
        const int base = cblk * 16;
        v2f Bv;
        Bv.x = Rm[(size_t)(kh * 2) * D_ + base + m];
        Bv.y = Rm[(size_t)(kh * 2 + 1) * D_ + base + m];
        v8f C;
#pragma unroll
        for (int j = 0; j < 8; ++j) {
            const int mm = j + kh * 8;
            C[j] = hb[(size_t)tok[mm] * D_ + base + m]; // unconditional; padded
        }                                               // rows get dl=0 -> D=C
        v8f Dv = __builtin_amdgcn_wmma_f32_16x16x4_f32(false, A2, false, Bv,
                                                       (short)0, C, false, false);
#pragma unroll
        for (int j = 0; j < 8; ++j) {
            const int mm = j + kh * 8;
            if (mm < cnt) out[outb + (size_t)tok[mm] * D_ + base + m] = Dv[j];
        }
    }
}

// ---------------------------------------------------------------------------
extern "C" void kernel_launch(void* const* d_in, const int* in_sizes, int n_in,
                              void* d_out, int out_size, void* d_ws, size_t ws_size,
                              hipStream_t stream)
{
    (void)in_sizes; (void)n_in; (void)out_size; (void)ws_size;

    const float* h   = (const float*)d_in[0];
    const MaskT* vis = (const MaskT*)d_in[1];
    const float* Rm  = (const float*)d_in[2];
    const float* Ww  = (const float*)d_in[3];
    const float* Wb  = (const float*)d_in[4];
    float* out = (float*)d_out;

    float* norms2 = (float*)d_ws;                 // B*S floats
    int* sidx = (int*)(norms2 + B_ * S_);         // B*MAXK ints
    int* scnt = sidx + B_ * MAXK;                 // B ints

    oi_norm_copy<<<B_ * S_ / 8, 256, 0, stream>>>(h, out, norms2);
    oi_select<<<B_, 256, 0, stream>>>(norms2, vis, sidx, scnt);
    oi_steer_wmma<<<B_ * TILES, 32, 0, stream>>>(h, Rm, Ww, Wb, sidx, scnt, out);
}